// SSIM_83829171683680
// MI455X (gfx1250) — compile-verified
//
#include <hip/hip_runtime.h>

// ---------------------------------------------------------------------------
// SSIM for B=16,C=3,H=W=512 fp32, 11x11 Gaussian (separable), VALID -> 502x502
// Strategy: per-wave 16x16 output tile; separable blur cast as band-matrix
// products through V_WMMA_F32_16X16X4_F32 (fp32 matrix path keeps the
// variance cancellation accurate). LDS-staged windows, deterministic 2-stage
// reduction.
// ---------------------------------------------------------------------------

typedef __attribute__((ext_vector_type(2))) float v2f;
typedef __attribute__((ext_vector_type(8))) float v8f;

#define IMG_H   512
#define IMG_W   512
#define OUT_HW  502            // 512 - 11 + 1
#define PLANES  48             // B*C
#define TILES_PER_PLANE (32 * 32)
#define TOTAL_TILES (PLANES * TILES_PER_PLANE)   // 49152
#define WAVES_PER_BLOCK 4
#define BLOCK_THREADS (WAVES_PER_BLOCK * 32)
#define NUM_BLOCKS (TOTAL_TILES / WAVES_PER_BLOCK)  // 12288

#define XS_STRIDE 28           // staged window: 32 rows x 28 cols
#define HS_STRIDE 20           // H scratch: 32 rows x 16 cols, stride 20 (bank spread)
#define SSIM_C1 1e-4f          // 0.01^2
#define SSIM_C2 4e-4f          // 0.02^2

__global__ __launch_bounds__(BLOCK_THREADS)
void ssim_tiles_kernel(const float* __restrict__ X,
                       const float* __restrict__ Y,
                       const float* __restrict__ Wk,
                       float* __restrict__ partial)
{
    __shared__ float gk[16];                                  // 1-D normalized Gaussian
    __shared__ float xs[WAVES_PER_BLOCK][32 * XS_STRIDE];
    __shared__ float ys[WAVES_PER_BLOCK][32 * XS_STRIDE];
    __shared__ float hs[WAVES_PER_BLOCK][32 * HS_STRIDE];
    __shared__ float bsum[WAVES_PER_BLOCK];

    const int tid  = threadIdx.x;
    // ghat_i = sum_j w[i][j]  (w = ghat ghat^T, sum ghat = 1); channel planes identical
    if (tid < 11) {
        float s = 0.f;
        #pragma unroll
        for (int j = 0; j < 11; ++j) s += Wk[tid * 11 + j];
        gk[tid] = s;
    }
    __syncthreads();

    const int wave = tid >> 5;
    const int lane = tid & 31;
    const int n    = lane & 15;   // N (or M) index within fragment
    const int hi   = lane >> 4;   // half-wave selector (K offset +2)

    // Band-matrix fragments: value = ghat[K - n], K = 4*kc + 2*hi + v.
    // Identical per-lane values serve as B in the horizontal pass and as A in
    // the vertical pass (f32 WMMA A/B lane layouts mirror each other).
    v2f gb[7];
    #pragma unroll
    for (int kc = 0; kc < 7; ++kc) {
        #pragma unroll
        for (int v = 0; v < 2; ++v) {
            int K = 4 * kc + 2 * hi + v;
            int d = K - n;
            gb[kc][v] = ((unsigned)d <= 10u) ? gk[d] : 0.f;
        }
    }

    // ---- tile decode -------------------------------------------------------
    const int tileId = blockIdx.x * WAVES_PER_BLOCK + wave;
    const int plane  = tileId >> 10;
    const int rem    = tileId & 1023;
    const int y0     = (rem >> 5) << 4;
    const int x0     = (rem & 31) << 4;

    const float* xp = X + (size_t)plane * (IMG_H * IMG_W);
    const float* yp = Y + (size_t)plane * (IMG_H * IMG_W);

    float* xw = xs[wave];
    float* yw = ys[wave];
    float* hw = hs[wave];

    // ---- stage 32x28 clamped windows of x and y into LDS -------------------
    if (lane < XS_STRIDE) {
        const int cc = min(x0 + lane, IMG_W - 1);
        #pragma unroll 4
        for (int r = 0; r < 32; ++r) {
            const int rr = min(y0 + r, IMG_H - 1);
            xw[r * XS_STRIDE + lane] = xp[rr * IMG_W + cc];
            yw[r * XS_STRIDE + lane] = yp[rr * IMG_W + cc];
        }
    }
    // per-wave LDS: DS ops are in-order within a wave, no barrier needed.

    // ---- 5 blurred fields: x, y, x^2, y^2, x*y -----------------------------
    v8f res[5];
    #pragma unroll
    for (int f = 0; f < 5; ++f) {
        // horizontal pass: two 16-row halves -> 32 rows of H in LDS
        #pragma unroll
        for (int h = 0; h < 2; ++h) {
            v8f acc = {0.f, 0.f, 0.f, 0.f, 0.f, 0.f, 0.f, 0.f};
            const int rowBase = (h * 16 + n) * XS_STRIDE + 2 * hi;
            #pragma unroll
            for (int kc = 0; kc < 7; ++kc) {
                const int off = rowBase + 4 * kc;
                v2f xv = *(const v2f*)(xw + off);
                v2f yv = *(const v2f*)(yw + off);
                v2f a  = (f == 0) ? xv
                       : (f == 1) ? yv
                       : (f == 2) ? xv * xv
                       : (f == 3) ? yv * yv
                                  : xv * yv;
                acc = __builtin_amdgcn_wmma_f32_16x16x4_f32(
                          false, a, false, gb[kc], (short)0, acc, false, false);
            }
            #pragma unroll
            for (int v = 0; v < 8; ++v)
                hw[(h * 16 + v + 8 * hi) * HS_STRIDE + n] = acc[v];
        }
        // vertical pass: Out = G^T x H
        v8f out = {0.f, 0.f, 0.f, 0.f, 0.f, 0.f, 0.f, 0.f};
        #pragma unroll
        for (int kc = 0; kc < 7; ++kc) {
            const int K = 4 * kc + 2 * hi;
            v2f b;
            b[0] = hw[K * HS_STRIDE + n];
            b[1] = hw[(K + 1) * HS_STRIDE + n];
            out = __builtin_amdgcn_wmma_f32_16x16x4_f32(
                      false, gb[kc], false, b, (short)0, out, false, false);
        }
        res[f] = out;
    }

    // ---- pointwise SSIM + masked partial sum -------------------------------
    float psum = 0.f;
    #pragma unroll
    for (int v = 0; v < 8; ++v) {
        const int oy = y0 + v + 8 * hi;
        const int ox = x0 + n;
        if (oy < OUT_HW && ox < OUT_HW) {
            const float mx = res[0][v], my = res[1][v];
            const float sxx = res[2][v] - mx * mx;
            const float syy = res[3][v] - my * my;
            const float sxy = res[4][v] - mx * my;
            const float l  = (2.f * mx * my + SSIM_C1) / (mx * mx + my * my + SSIM_C1);
            const float cs = (2.f * sxy + SSIM_C2) / (sxx + syy + SSIM_C2);
            psum += l * cs;
        }
    }
    #pragma unroll
    for (int off = 16; off > 0; off >>= 1)
        psum += __shfl_down(psum, off, 32);
    if (lane == 0) bsum[wave] = psum;
    __syncthreads();
    if (tid == 0)
        partial[blockIdx.x] = bsum[0] + bsum[1] + bsum[2] + bsum[3];
}

__global__ __launch_bounds__(256)
void ssim_reduce_kernel(const float* __restrict__ partial, int nPart,
                        float* __restrict__ out)
{
    __shared__ double s[256];
    const int tid = threadIdx.x;
    double acc = 0.0;
    for (int i = tid; i < nPart; i += 256) acc += (double)partial[i];
    s[tid] = acc;
    __syncthreads();
    #pragma unroll
    for (int step = 128; step > 0; step >>= 1) {
        if (tid < step) s[tid] += s[tid + step];
        __syncthreads();
    }
    if (tid == 0) out[0] = 1.f - (float)(s[0] / (double)PLANES);
}

extern "C" void kernel_launch(void* const* d_in, const int* in_sizes, int n_in,
                              void* d_out, int out_size, void* d_ws, size_t ws_size,
                              hipStream_t stream)
{
    const float* x = (const float*)d_in[0];
    const float* y = (const float*)d_in[1];
    const float* w = (const float*)d_in[2];
    float* out     = (float*)d_out;
    float* partial = (float*)d_ws;   // NUM_BLOCKS floats

    ssim_tiles_kernel<<<NUM_BLOCKS, BLOCK_THREADS, 0, stream>>>(x, y, w, partial);
    ssim_reduce_kernel<<<1, 256, 0, stream>>>(partial, NUM_BLOCKS, out);
}